// GAT_17428977287427
// MI455X (gfx1250) — compile-verified
//
#include <hip/hip_runtime.h>
#include <hip/hip_bf16.h>

// ---------------- problem constants (match reference) ----------------
#define NNODES 50000
#define NEDGES 800000
#define E2     (NEDGES + NNODES)   // edges + self loops = 850000
#define FIN    128
#define H1     4                   // heads layer 1
#define CCH    64                  // channels per head
#define HC1    (H1 * CCH)          // 256
#define HID    64
#define NG     64                  // graphs

typedef float v2f __attribute__((ext_vector_type(2)));
typedef float v8f __attribute__((ext_vector_type(8)));
typedef int   ai4 __attribute__((vector_size(16)));   // int __vector(4)

// ---------------- WMMA fp32 GEMM: C[M,Nc] = A[M,K] @ B[K,Nc] ----------------
// Block = 256 threads = 8 waves; block covers 8 row-tiles x 1 col-tile.
// B[K,16] strip staged in LDS once per block (async-to-LDS when available),
// then one V_WMMA_F32_16X16X4_F32 per wave per 4-wide K step.
template <int K>
__global__ __launch_bounds__(256) void wmma_gemm_f32(
    const float* __restrict__ A, const float* __restrict__ B,
    float* __restrict__ C, int M, int Nc)
{
    __shared__ float Blds[K * 16];          // k-major: Blds[k*16 + col]

    const int lane = threadIdx.x;           // 0..31
    const int wv   = threadIdx.y;           // 0..7
    const int tid  = wv * 32 + lane;
    const int n0   = blockIdx.y * 16;       // column tile origin

    // ---- cooperative stage of B[K, n0:n0+16] into LDS ----
#if __has_builtin(__builtin_amdgcn_global_load_async_to_lds_b128)
    // one b128 chunk = 4 floats; K*4 chunks total
    for (int c = tid; c < K * 4; c += 256) {
        float* gsrc = (float*)B + (size_t)(c >> 2) * Nc + n0 + (c & 3) * 4;
        __builtin_amdgcn_global_load_async_to_lds_b128(
            (__attribute__((address_space(1))) ai4*)gsrc,
            (__attribute__((address_space(3))) ai4*)&Blds[c * 4],
            0, 0);
    }
    asm volatile("s_wait_asynccnt 0x0" ::: "memory");
#else
    for (int i = tid; i < K * 16; i += 256)
        Blds[i] = B[(size_t)(i >> 4) * Nc + n0 + (i & 15)];
#endif
    __syncthreads();

    const int tileM = blockIdx.x * 8 + wv;
    if (tileM * 16 < M) {                   // whole wave predicate (EXEC all-ones)
        const int row = lane & 15;
        const int kk  = (lane >> 4) << 1;   // 0 for lanes 0-15, 2 for 16-31
        const float* Arow = A + (size_t)(tileM * 16 + row) * K;
        const float* Bl   = &Blds[row];     // row == lane&15 == column index

        v8f acc = {0.f, 0.f, 0.f, 0.f, 0.f, 0.f, 0.f, 0.f};
#pragma unroll 8
        for (int k = 0; k < K; k += 4) {
            v2f a = *(const v2f*)(Arow + k + kk);      // A[row, k+kk .. k+kk+1]
            v2f b;
            b.x = Bl[(k + kk) * 16];                   // B[k+kk,   col] from LDS
            b.y = Bl[(k + kk + 1) * 16];               // B[k+kk+1, col] from LDS
            acc = __builtin_amdgcn_wmma_f32_16x16x4_f32(
                false, a, false, b, (short)0, acc, false, false);
        }

        // C layout: vgpr r -> M = r (lanes 0-15) / r+8 (lanes 16-31), N = lane&15
        const int crow0 = tileM * 16 + (lane >> 4) * 8;
        float* Cp = C + (size_t)crow0 * Nc + n0 + (lane & 15);
#pragma unroll
        for (int r = 0; r < 8; ++r)
            Cp[(size_t)r * Nc] = acc[r];
    }
}

// ---------------- helpers ----------------
__device__ inline void edge_nodes(const int* __restrict__ ei, int e, int& s, int& d)
{
    if (e < NEDGES) { s = ei[e]; d = ei[NEDGES + e]; }
    else            { s = e - NEDGES; d = s; }      // self loop
}

__device__ inline void atomicMaxFloat(float* addr, float val)
{
    int* ia = (int*)addr;
    int old = __float_as_int(*addr);
    while (__int_as_float(old) < val) {
        int assumed = old;
        old = atomicCAS(ia, assumed, __float_as_int(val));
        if (old == assumed) break;
    }
}

__global__ void fill_kernel(float* __restrict__ p, int n, float v)
{
    int i = blockIdx.x * blockDim.x + threadIdx.x;
    if (i < n) p[i] = v;
}

// ---------------- per-(node,head) attention projections s,d ----------------
template <int H>
__global__ void sd_kernel(const float* __restrict__ h,
                          const float* __restrict__ asrc,
                          const float* __restrict__ adst,
                          float* __restrict__ s, float* __restrict__ d)
{
    int t = blockIdx.x * blockDim.x + threadIdx.x;
    if (t >= NNODES * H) return;
    int node = t / H, hd = t - node * H;
    const float4* hp = (const float4*)(h + (size_t)node * H * CCH + hd * CCH);
    const float4* as = (const float4*)(asrc + hd * CCH);
    const float4* ad = (const float4*)(adst + hd * CCH);
    float ss = 0.f, dd = 0.f;
#pragma unroll 4
    for (int c = 0; c < CCH / 4; ++c) {
        float4 v = hp[c], a = as[c], b = ad[c];
        ss += v.x * a.x + v.y * a.y + v.z * a.z + v.w * a.w;
        dd += v.x * b.x + v.y * b.y + v.z * b.z + v.w * b.w;
    }
    s[t] = ss; d[t] = dd;
}

// ---------------- edge pass 1: leaky-relu score + segment max ----------------
template <int H>
__global__ void edge_max_kernel(const int* __restrict__ ei,
                                const float* __restrict__ s,
                                const float* __restrict__ d,
                                float* __restrict__ e_out,
                                float* __restrict__ m)
{
    int t = blockIdx.x * blockDim.x + threadIdx.x;
    if (t >= E2 * H) return;
    int eid = t / H, hd = t - eid * H;
    int sn, dn; edge_nodes(ei, eid, sn, dn);
    float v = s[sn * H + hd] + d[dn * H + hd];
    v = v > 0.f ? v : 0.2f * v;              // leaky relu, slope 0.2
    e_out[t] = v;
    atomicMaxFloat(&m[dn * H + hd], v);
}

// ---------------- edge pass 2: p = exp(e - m[dst]); z[dst] += p ----------------
template <int H>
__global__ void edge_expsum_kernel(const int* __restrict__ ei,
                                   float* __restrict__ e_out,
                                   const float* __restrict__ m,
                                   float* __restrict__ z)
{
    int t = blockIdx.x * blockDim.x + threadIdx.x;
    if (t >= E2 * H) return;
    int eid = t / H, hd = t - eid * H;
    int sn, dn; edge_nodes(ei, eid, sn, dn);
    float p = expf(e_out[t] - m[dn * H + hd]);
    e_out[t] = p;                            // store numerator
    atomicAdd(&z[dn * H + hd], p);
}

// ---------------- edge pass 3: agg[dst] += alpha * h[src] ----------------
// block = one edge, blockDim.x = H*CCH
template <int H>
__global__ void aggregate_kernel(const int* __restrict__ ei,
                                 const float* __restrict__ hfeat,
                                 const float* __restrict__ p,
                                 const float* __restrict__ z,
                                 float* __restrict__ agg)
{
    int eid = blockIdx.x;
    int t   = threadIdx.x;                   // 0 .. H*CCH-1
    int hd  = t >> 6;
    int sn, dn; edge_nodes(ei, eid, sn, dn);
    float alpha = p[(size_t)eid * H + hd] / z[dn * H + hd];
    atomicAdd(&agg[(size_t)dn * (H * CCH) + t],
              alpha * hfeat[(size_t)sn * (H * CCH) + t]);
}

// ---------------- bias + ELU (in place) ----------------
__global__ void bias_elu_kernel(float* __restrict__ a, const float* __restrict__ b,
                                int total, int HC)
{
    int i = blockIdx.x * blockDim.x + threadIdx.x;
    if (i >= total) return;
    float v = a[i] + b[i % HC];
    a[i] = v > 0.f ? v : (expf(v) - 1.f);
}

// ---------------- global mean pool (sum + count via atomics) ----------------
__global__ void pool_kernel(const float* __restrict__ a2, const int* __restrict__ batch,
                            float* __restrict__ pooled, float* __restrict__ cnt)
{
    int t = blockIdx.x * blockDim.x + threadIdx.x;
    if (t >= NNODES * HID) return;
    int node = t >> 6, c = t & 63;
    int g = batch[node];
    atomicAdd(&pooled[g * HID + c], a2[t]);
    if (c == 0) atomicAdd(&cnt[g], 1.0f);
}

// ---------------- final two linear heads ----------------
__global__ void head_kernel(const float* __restrict__ pooled, const float* __restrict__ cnt,
                            const float* __restrict__ Wm, const float* __restrict__ bm,
                            const float* __restrict__ Wt, const float* __restrict__ bt,
                            float* __restrict__ out)
{
    int g = threadIdx.x;
    if (g >= NG) return;
    float c = cnt[g]; c = c > 1.f ? c : 1.f;
    float sm = 0.f, st = 0.f;
    for (int i = 0; i < HID; ++i) {
        float v = pooled[g * HID + i] / c;
        sm += v * Wm[i];
        st += v * Wt[i];
    }
    out[g]      = sm + bm[0];
    out[NG + g] = st + bt[0];
}

// ---------------- launcher ----------------
extern "C" void kernel_launch(void* const* d_in, const int* in_sizes, int n_in,
                              void* d_out, int out_size, void* d_ws, size_t ws_size,
                              hipStream_t stream)
{
    (void)in_sizes; (void)n_in; (void)out_size; (void)ws_size;

    const float* x     = (const float*)d_in[0];
    const int*   ei    = (const int*)d_in[1];   // [2,E] flat: src then dst
    const int*   batch = (const int*)d_in[2];
    const float* W1    = (const float*)d_in[3];
    const float* as1   = (const float*)d_in[4];
    const float* ad1   = (const float*)d_in[5];
    const float* b1    = (const float*)d_in[6];
    const float* W2    = (const float*)d_in[7];
    const float* as2   = (const float*)d_in[8];
    const float* ad2   = (const float*)d_in[9];
    const float* b2    = (const float*)d_in[10];
    const float* Wm    = (const float*)d_in[11];
    const float* bm    = (const float*)d_in[12];
    const float* Wt    = (const float*)d_in[13];
    const float* bt    = (const float*)d_in[14];
    float* out = (float*)d_out;

    // ---- workspace layout (floats) ----
    float* ws = (float*)d_ws;
    size_t o = 0;
    float* h1   = ws + o; o += (size_t)NNODES * HC1;   // 12.8M
    float* agg1 = ws + o; o += (size_t)NNODES * HC1;   // 12.8M (becomes a1)
    float* h2   = ws + o; o += (size_t)NNODES * HID;   // 3.2M
    float* agg2 = ws + o; o += (size_t)NNODES * HID;   // 3.2M (becomes a2)
    float* s1   = ws + o; o += (size_t)NNODES * H1;
    float* d1   = ws + o; o += (size_t)NNODES * H1;
    float* m1   = ws + o; o += (size_t)NNODES * H1;
    float* z1   = ws + o; o += (size_t)NNODES * H1;
    float* s2   = ws + o; o += (size_t)NNODES;
    float* d2   = ws + o; o += (size_t)NNODES;
    float* m2   = ws + o; o += (size_t)NNODES;
    float* z2   = ws + o; o += (size_t)NNODES;
    float* ebuf = ws + o; o += (size_t)E2 * H1;        // reused for layer 2
    float* pooled = ws + o; o += (size_t)NG * HID;
    float* cnt    = ws + o; o += (size_t)NG;

    // ---- init scratch (graph-capture-safe memsets + fill) ----
    (void)hipMemsetAsync(agg1, 0, (size_t)NNODES * HC1 * sizeof(float), stream);
    (void)hipMemsetAsync(agg2, 0, (size_t)NNODES * HID * sizeof(float), stream);
    (void)hipMemsetAsync(z1,   0, (size_t)NNODES * H1 * sizeof(float), stream);
    (void)hipMemsetAsync(z2,   0, (size_t)NNODES * sizeof(float), stream);
    (void)hipMemsetAsync(pooled, 0, (size_t)(NG * HID + NG) * sizeof(float), stream);
    fill_kernel<<<(NNODES * H1 + 255) / 256, 256, 0, stream>>>(m1, NNODES * H1, -1e30f);
    fill_kernel<<<(NNODES + 255) / 256, 256, 0, stream>>>(m2, NNODES, -1e30f);

    const int mblocks = (NNODES / 16 + 7) / 8;         // 391 row-tile groups

    // ---- layer 1: h1 = x @ W1  (50000x128 @ 128x256) ----
    wmma_gemm_f32<FIN><<<dim3(mblocks, HC1 / 16), dim3(32, 8), 0, stream>>>(
        x, W1, h1, NNODES, HC1);
    sd_kernel<H1><<<(NNODES * H1 + 255) / 256, 256, 0, stream>>>(h1, as1, ad1, s1, d1);
    edge_max_kernel<H1><<<(E2 * H1 + 255) / 256, 256, 0, stream>>>(ei, s1, d1, ebuf, m1);
    edge_expsum_kernel<H1><<<(E2 * H1 + 255) / 256, 256, 0, stream>>>(ei, ebuf, m1, z1);
    aggregate_kernel<H1><<<E2, H1 * CCH, 0, stream>>>(ei, h1, ebuf, z1, agg1);
    bias_elu_kernel<<<((NNODES * HC1) + 255) / 256, 256, 0, stream>>>(
        agg1, b1, NNODES * HC1, HC1);

    // ---- layer 2: h2 = a1 @ W2  (50000x256 @ 256x64), heads=1 ----
    wmma_gemm_f32<HC1><<<dim3(mblocks, HID / 16), dim3(32, 8), 0, stream>>>(
        agg1, W2, h2, NNODES, HID);
    sd_kernel<1><<<(NNODES + 255) / 256, 256, 0, stream>>>(h2, as2, ad2, s2, d2);
    edge_max_kernel<1><<<(E2 + 255) / 256, 256, 0, stream>>>(ei, s2, d2, ebuf, m2);
    edge_expsum_kernel<1><<<(E2 + 255) / 256, 256, 0, stream>>>(ei, ebuf, m2, z2);
    aggregate_kernel<1><<<E2, CCH, 0, stream>>>(ei, h2, ebuf, z2, agg2);
    bias_elu_kernel<<<((NNODES * HID) + 255) / 256, 256, 0, stream>>>(
        agg2, b2, NNODES * HID, HID);

    // ---- pool + heads ----
    pool_kernel<<<((NNODES * HID) + 255) / 256, 256, 0, stream>>>(agg2, batch, pooled, cnt);
    head_kernel<<<1, 64, 0, stream>>>(pooled, cnt, Wm, bm, Wt, bt, out);
}